// REGIN_9878424781413
// MI455X (gfx1250) — compile-verified
//
#include <hip/hip_runtime.h>

typedef __attribute__((ext_vector_type(2))) float v2f;
typedef __attribute__((ext_vector_type(8))) float v8f;

constexpr int N0 = 50000, N1 = 30000, N2 = 20000;
constexpr int NTOT = N0 + N1 + N2;   // 100000
constexpr int E    = 1600000;
constexpr int D    = 128;
constexpr int DOUT = 64;

// ---------------------------------------------------------------------------
// GEMM: C[M, NOUT] = act(A[M,K] @ W[NOUT,K]^T + bias) via V_WMMA_F32_16X16X4_F32
// One wave -> one 16x16 tile; wave w owns columns [w*16, w*16+16);
// blockIdx.x owns rows [bx*16, bx*16+16). All M are multiples of 16, so EXEC
// stays all-ones (WMMA requirement) and no bounds checks are needed.
//
// f32 16x16x4 operand layout (ISA 7.12.2):
//   A (16x4): lane<16 -> row=lane,   {K=k,K=k+1}; lane>=16 -> row=lane-16, {K=k+2,K=k+3}
//   B (4x16): same pattern over W rows (W row-major [NOUT,K]; B = W^T tile)
//   C/D:      vgpr v, lane<16 -> C[m0+v][n0+lane]; lane>=16 -> C[m0+8+v][n0+lane-16]
// ---------------------------------------------------------------------------
template<int K, int NOUT, bool BIAS, bool RELU>
__global__ __launch_bounds__(256)
void gemm_wmma(const float* __restrict__ A, const float* __restrict__ W,
               const float* __restrict__ bias, float* __restrict__ C)
{
    const int lane = threadIdx.x & 31;
    const int wave = threadIdx.x >> 5;
    const int m0   = blockIdx.x << 4;
    const int n0   = wave << 4;
    const int r    = lane & 15;             // row within A/W tile for this lane
    const int koff = (lane >> 4) << 1;      // lanes 16..31 handle K+2,K+3

    const float* ap = A + (m0 + r) * K + koff;
    const float* wp = W + (n0 + r) * K + koff;

    v8f c0 = {}, c1 = {};
#pragma unroll 4
    for (int k = 0; k < K; k += 8) {
        v2f a0 = *(const v2f*)(ap + k);
        v2f b0 = *(const v2f*)(wp + k);
        v2f a1 = *(const v2f*)(ap + k + 4);
        v2f b1 = *(const v2f*)(wp + k + 4);
        c0 = __builtin_amdgcn_wmma_f32_16x16x4_f32(false, a0, false, b0,
                                                   (short)0, c0, false, false);
        c1 = __builtin_amdgcn_wmma_f32_16x16x4_f32(false, a1, false, b1,
                                                   (short)0, c1, false, false);
    }
    v8f c = c0 + c1;

    float bv = 0.0f;
    if (BIAS) bv = bias[n0 + r];

    const int rowBase = m0 + ((lane >> 4) << 3);   // +8 for upper half-wave
    float* cp = C + rowBase * NOUT + n0 + r;
#pragma unroll
    for (int v = 0; v < 8; ++v) {
        float x = c[v] + bv;
        if (RELU) x = fmaxf(x, 0.0f);
        cp[v * NOUT] = x;
    }
}

// Native no-return fp32 atomic add, device scope (cross-WGP accumulation).
__device__ __forceinline__ void atom_add_f32(float* p, float v)
{
    asm volatile("global_atomic_add_f32 %0, %1, off scope:SCOPE_DEV"
                 :: "v"(p), "v"(v) : "memory");
}

// ---------------------------------------------------------------------------
// Edge aggregation: agg[dst] += et_w[e_feat] * h[src].  One wave per edge.
// Edge index forced uniform via readfirstlane so src/dst/etype/weight become
// SGPR s_loads (KMcnt path); each lane moves one float4 of the 128-float row
// and issues 4 native global_atomic_add_f32. h (51MB) and agg (51MB) both fit
// in the 192MB L2, so this stage is L2-bandwidth/atomic bound, not HBM bound.
// E % 8 == 0 -> grid is exact, no bounds check, EXEC all-ones.
// ---------------------------------------------------------------------------
__global__ __launch_bounds__(256)
void edge_agg(const float* __restrict__ h, const int* __restrict__ src,
              const int* __restrict__ dst, const int* __restrict__ ef,
              const float* __restrict__ etw, float* __restrict__ agg)
{
    const int lane = threadIdx.x & 31;
    const int e = __builtin_amdgcn_readfirstlane(blockIdx.x * 8 + (threadIdx.x >> 5));

    const int   s = src[e];
    const int   d = dst[e];
    const float w = etw[ef[e]];

    const float4 v = *(const float4*)(h + s * D + lane * 4);
    float* o = agg + d * D + lane * 4;
    atom_add_f32(o + 0, w * v.x);
    atom_add_f32(o + 1, w * v.y);
    atom_add_f32(o + 2, w * v.z);
    atom_add_f32(o + 3, w * v.w);
}

__global__ __launch_bounds__(256)
void zero_f4(float4* __restrict__ p, long n4)
{
    long i = (long)blockIdx.x * blockDim.x + threadIdx.x;
    if (i < n4) p[i] = make_float4(0.f, 0.f, 0.f, 0.f);
}

// z = (1+eps)*h + agg   (may run in-place: z aliases agg, pure elementwise)
__global__ __launch_bounds__(256)
void gin_combine(const float4* h, const float4* agg, const float* eps_p,
                 float4* z, long n4)
{
    long i = (long)blockIdx.x * blockDim.x + threadIdx.x;
    if (i >= n4) return;
    float s  = 1.0f + *eps_p;
    float4 a = h[i];
    float4 g = agg[i];
    z[i] = make_float4(fmaf(s, a.x, g.x), fmaf(s, a.y, g.y),
                       fmaf(s, a.z, g.z), fmaf(s, a.w, g.w));
}

extern "C" void kernel_launch(void* const* d_in, const int* in_sizes, int n_in,
                              void* d_out, int out_size, void* d_ws, size_t ws_size,
                              hipStream_t stream)
{
    const float* feat0  = (const float*)d_in[0];
    const float* feat1  = (const float*)d_in[1];
    const float* feat2  = (const float*)d_in[2];
    const float* W0     = (const float*)d_in[3];
    const float* b0     = (const float*)d_in[4];
    const float* W1     = (const float*)d_in[5];
    const float* b1     = (const float*)d_in[6];
    const float* W2     = (const float*)d_in[7];
    const float* b2     = (const float*)d_in[8];
    const float* et_w0  = (const float*)d_in[9];
    const float* eps0   = (const float*)d_in[10];
    const float* mlp0_W = (const float*)d_in[11];
    const float* et_w1  = (const float*)d_in[12];
    const float* eps1   = (const float*)d_in[13];
    const float* out_W  = (const float*)d_in[14];
    const int*   src    = (const int*)d_in[15];
    const int*   dst    = (const int*)d_in[16];
    const int*   ef     = (const int*)d_in[17];

    float* out  = (float*)d_out;                  // [NTOT, 64]
    float* hout = out + (long)NTOT * DOUT;        // [NTOT, 128]

    float* buf_h   = (float*)d_ws;                // [NTOT, 128]
    float* buf_agg = buf_h + (long)NTOT * D;      // [NTOT, 128]

    const long n4      = (long)NTOT * D / 4;      // 3.2M float4
    const int  zgrid   = (int)((n4 + 255) / 256);
    const int  ew_grid = E / 8;                   // 8 waves/block, 1 edge/wave

    // 1) per-type projections -> buf_h  (bias, no act)
    gemm_wmma<256, 128, true, false><<<N0 / 16, 256, 0, stream>>>(feat0, W0, b0, buf_h);
    gemm_wmma<128, 128, true, false><<<N1 / 16, 256, 0, stream>>>(feat1, W1, b1,
                                                                  buf_h + N0 * D);
    gemm_wmma< 64, 128, true, false><<<N2 / 16, 256, 0, stream>>>(feat2, W2, b2,
                                                                  buf_h + (N0 + N1) * D);

    // 2) layer-1 aggregation
    zero_f4<<<zgrid, 256, 0, stream>>>((float4*)buf_agg, n4);
    edge_agg<<<ew_grid, 256, 0, stream>>>(buf_h, src, dst, ef, et_w0, buf_agg);
    // z = (1+eps0)*h + agg, in-place into buf_agg
    gin_combine<<<zgrid, 256, 0, stream>>>(
        (const float4*)buf_h, (const float4*)buf_agg, eps0, (float4*)buf_agg, n4);
    // h1 = relu(z @ mlp0_W^T) -> buf_h (overwrite)
    gemm_wmma<128, 128, false, true><<<NTOT / 16, 256, 0, stream>>>(buf_agg, mlp0_W,
                                                                    nullptr, buf_h);

    // 3) layer-2 aggregation (no MLP, no act)
    zero_f4<<<zgrid, 256, 0, stream>>>((float4*)buf_agg, n4);
    edge_agg<<<ew_grid, 256, 0, stream>>>(buf_h, src, dst, ef, et_w1, buf_agg);
    // final h = (1+eps1)*h1 + agg -> d_out h region
    gin_combine<<<zgrid, 256, 0, stream>>>(
        (const float4*)buf_h, (const float4*)buf_agg, eps1, (float4*)hout, n4);

    // 4) out = h @ out_W^T -> d_out out region (NOUT=64 -> 4 waves/block)
    gemm_wmma<128, 64, false, false><<<NTOT / 16, 128, 0, stream>>>(hout, out_W,
                                                                    nullptr, out);
}